// CustomPolicyAndValue_16655883174593
// MI455X (gfx1250) — compile-verified
//
#include <hip/hip_runtime.h>
#include <hip/hip_bf16.h>

// ---------------------------------------------------------------------------
// GAT policy+value network for MI455X (gfx1250, wave32).
//  * Dense GEMMs (80000x128 @ 128x128) on matrix cores: V_WMMA_F32_16X16X4_F32
//    (fp32 precision; the pipeline is HBM/L2 bound so low-precision buys
//    nothing and would cost accuracy through the softmax).
//  * W (64KB) staged into LDS once per block via the CDNA5 async copy path
//    (global_load_async_to_lds_b128 + s_wait_asynccnt), read back conflict-
//    free via ds_load_b64 (row pitch 132 floats -> banks stride 4).
//  * Each wave computes a 16x128 output stripe: A fragment loaded once per
//    k-step, reused for 8 WMMAs -> matrix pipe not gated on VMEM waits.
//  * Edge softmax/aggregation: L2-resident f32 atomics (h = 41MB < 192MB L2).
// ---------------------------------------------------------------------------

typedef __attribute__((ext_vector_type(2))) float v2f;
typedef __attribute__((ext_vector_type(8))) float v8f;

#define CDIM 128
#define NEG_SLOPE 0.2f
#define LDSPITCH 132                       // floats per W row in LDS (16B aligned, bank stride 4)
#define GEMM_LDS_BYTES (CDIM * LDSPITCH * 4)  // 67584 B ( < 320KB WGP LDS )

// ---------------- order-preserving float <-> uint for atomic max ------------
__device__ __forceinline__ unsigned fenc(float f) {
    unsigned u = __float_as_uint(f);
    return (u & 0x80000000u) ? ~u : (u | 0x80000000u);
}
__device__ __forceinline__ float fdec(unsigned u) {
    return (u & 0x80000000u) ? __uint_as_float(u & 0x7FFFFFFFu)
                             : __uint_as_float(~u);
}

// ---------------- WMMA fp32 GEMM: D[M,128] = A[M,128] @ W^T[128,128] --------
// Block = 256 threads = 8 waves. Block b: wave w computes output rows
// [16*(8b+w), 16*(8b+w)+16) x all 128 cols (8 accumulators).
// fp32 WMMA fragment layout (ISA 7.12.2):
//   A 16x4 : lane 0-15 -> M=lane, K={k0,k0+1}; lane 16-31 -> M=lane-16, K={k0+2,k0+3}
//   B 4x16 : lane 0-15 -> N=lane, K={k0,k0+1}; lane 16-31 -> N=lane-16, K={k0+2,k0+3}
//   (B[k][n] = W[n][k] -> B fragment = 2 consecutive elements of W row n)
//   D 16x16: VGPR v -> M = v + 8*(lane>=16), N = lane&15
__global__ __launch_bounds__(256) void gemm_wmma_f32(
    const float* __restrict__ A, const float* __restrict__ W,
    float* __restrict__ D, int M)
{
    extern __shared__ float lw[];   // dynamic LDS starts at offset 0 (no statics)

    const int wave = threadIdx.x >> 5;
    const int lane = threadIdx.x & 31;
    const int tm   = blockIdx.x * 8 + wave;      // 16-row tile index
    const bool active = (tm * 16 < M);

    // ---- async copy W (128x128 f32) into LDS, padded pitch 132 floats ----
    // 4096 chunks of 16B; each wave issues 16 async-to-LDS instructions.
    for (int i = threadIdx.x; i < CDIM * (CDIM / 4); i += 256) {
        const int row = i >> 5;           // 32 chunks per row
        const int c   = i & 31;           // 16B chunk within row
        const float* g = W + (size_t)row * CDIM + c * 4;
        unsigned ldsOff = (unsigned)((row * LDSPITCH + c * 4) * 4);  // bytes
        asm volatile("global_load_async_to_lds_b128 %0, %1, off"
                     :: "v"(ldsOff), "v"(g)
                     : "memory");
    }
    asm volatile("s_wait_asynccnt 0x0" ::: "memory");
    __syncthreads();

    if (!active) return;   // wave-uniform (grid is exact for M=80000 anyway)

    const int r     = lane & 15;
    const int khalf = lane >> 4;     // 0/1 -> K offset 0/2

    const float* __restrict__ arow = A + (size_t)(tm * 16 + r) * CDIM;

    v8f acc[8] = {};
#pragma unroll 4
    for (int k0 = 0; k0 < CDIM; k0 += 4) {
        const int k = k0 + khalf * 2;
        v2f a = *(const v2f*)(arow + k);              // A[m][k..k+1]  (global)
#pragma unroll
        for (int tn = 0; tn < 8; ++tn) {              // 8 col tiles from LDS
            v2f b = *(const v2f*)(lw + (tn * 16 + r) * LDSPITCH + k);
            acc[tn] = __builtin_amdgcn_wmma_f32_16x16x4_f32(
                false, a, false, b, (short)0, acc[tn], false, false);
        }
    }

    const int rbase = tm * 16 + khalf * 8;
#pragma unroll
    for (int tn = 0; tn < 8; ++tn) {
        const int col = tn * 16 + r;
#pragma unroll
        for (int v = 0; v < 8; ++v)
            D[(size_t)(rbase + v) * CDIM + col] = acc[tn][v];
    }
}

// ---------------- per-node attention dot products ---------------------------
__global__ void node_dots(const float* __restrict__ h,
                          const float* __restrict__ att_src,
                          const float* __restrict__ att_dst,
                          float* __restrict__ as_, float* __restrict__ ad_, int M)
{
    int n = blockIdx.x * blockDim.x + threadIdx.x;
    if (n >= M) return;
    const float4* hp = (const float4*)(h + (size_t)n * CDIM);
    const float4* ap = (const float4*)att_src;
    const float4* bp = (const float4*)att_dst;
    float s = 0.f, d = 0.f;
#pragma unroll
    for (int i = 0; i < CDIM / 4; ++i) {
        float4 v = hp[i], a = ap[i], b = bp[i];
        s += v.x * a.x + v.y * a.y + v.z * a.z + v.w * a.w;
        d += v.x * b.x + v.y * b.y + v.z * b.z + v.w * b.w;
    }
    as_[n] = s;
    ad_[n] = d;
}

// ---------------- small dot (k_e = We . att_edge), n <= 128 -----------------
__global__ void dot_small(const float* __restrict__ a, const float* __restrict__ b,
                          int n, float* __restrict__ out)
{
    __shared__ float sm[128];
    int t = threadIdx.x;
    sm[t] = (t < n) ? a[t] * b[t] : 0.f;
    __syncthreads();
    for (int off = 64; off > 0; off >>= 1) {
        if (t < off) sm[t] += sm[t + off];
        __syncthreads();
    }
    if (t == 0) *out = sm[0];
}

// ---------------- grid-stride sum (edge_attr mean numerator) ----------------
__global__ void sum_reduce(const float* __restrict__ a, int n, float* __restrict__ out)
{
    __shared__ float sm[256];
    float s = 0.f;
    for (int i = blockIdx.x * blockDim.x + threadIdx.x; i < n;
         i += gridDim.x * blockDim.x)
        s += a[i];
    sm[threadIdx.x] = s;
    __syncthreads();
    for (int off = 128; off > 0; off >>= 1) {
        if (threadIdx.x < off) sm[threadIdx.x] += sm[threadIdx.x + off];
        __syncthreads();
    }
    if (threadIdx.x == 0) atomicAdd(out, sm[0]);
}

// ---------------- edge score + segment max ----------------------------------
__global__ void edge_score(const int* __restrict__ src, const int* __restrict__ dst,
                           const float* __restrict__ ea, int E0, int Etot,
                           const float* __restrict__ as_, const float* __restrict__ ad_,
                           const float* __restrict__ ke_p, const float* __restrict__ easum_p,
                           float* __restrict__ score, unsigned* __restrict__ mx)
{
    int e = blockIdx.x * blockDim.x + threadIdx.x;
    if (e >= Etot) return;
    int s, d;
    float eav;
    if (e < E0) { s = src[e]; d = dst[e]; eav = ea[e]; }
    else        { s = d = e - E0; eav = (*easum_p) * (1.0f / (float)E0); }
    float x = as_[s] + ad_[d] + eav * (*ke_p);
    x = (x > 0.f) ? x : NEG_SLOPE * x;   // leaky relu
    score[e] = x;
    atomicMax(&mx[d], fenc(x));
}

// ---------------- exp(score - max) + segment sum ----------------------------
__global__ void edge_exp(const int* __restrict__ dst, int E0, int Etot,
                         float* __restrict__ score, const unsigned* __restrict__ mx,
                         float* __restrict__ den)
{
    int e = blockIdx.x * blockDim.x + threadIdx.x;
    if (e >= Etot) return;
    int d = (e < E0) ? dst[e] : e - E0;
    float ex = expf(score[e] - fdec(mx[d]));
    score[e] = ex;
    atomicAdd(&den[d], ex);
}

// ---------------- weighted aggregation, C = 128 -----------------------------
// one wave per edge; each lane handles 4 channels (32*4 = 128)
__global__ __launch_bounds__(256) void edge_agg128(
    const int* __restrict__ src, const int* __restrict__ dst, int E0, int Etot,
    const float* __restrict__ ex, const float* __restrict__ den,
    const float* __restrict__ h, float* __restrict__ out)
{
    int e = blockIdx.x * 8 + (int)(threadIdx.x >> 5);
    if (e >= Etot) return;
    int lane = threadIdx.x & 31;
    int s, d;
    if (e < E0) { s = src[e]; d = dst[e]; } else { s = d = e - E0; }
    float alpha = ex[e] / den[d];
    float4 hv = ((const float4*)(h + (size_t)s * CDIM))[lane];
    float* o = out + (size_t)d * CDIM + lane * 4;
    atomicAdd(o + 0, alpha * hv.x);
    atomicAdd(o + 1, alpha * hv.y);
    atomicAdd(o + 2, alpha * hv.z);
    atomicAdd(o + 3, alpha * hv.w);
}

// ---------------- bias + relu (between layers) ------------------------------
__global__ void post_act(float* __restrict__ o, const float* __restrict__ bias, int total)
{
    int i = blockIdx.x * blockDim.x + threadIdx.x;
    if (i >= total) return;
    float v = o[i] + bias[i & (CDIM - 1)];
    o[i] = v > 0.f ? v : 0.f;
}

// ---------------- last layer: matvec 128 -> 1 + attention scalars -----------
__global__ void matvec_last(const float* __restrict__ hin, const float* __restrict__ W,
                            const float* __restrict__ att_src, const float* __restrict__ att_dst,
                            float* __restrict__ h1, float* __restrict__ as_,
                            float* __restrict__ ad_, int M)
{
    int n = blockIdx.x * blockDim.x + threadIdx.x;
    if (n >= M) return;
    const float4* hp = (const float4*)(hin + (size_t)n * CDIM);
    const float4* wp = (const float4*)W;
    float s = 0.f;
#pragma unroll
    for (int i = 0; i < CDIM / 4; ++i) {
        float4 v = hp[i], w = wp[i];
        s += v.x * w.x + v.y * w.y + v.z * w.z + v.w * w.w;
    }
    h1[n] = s;
    as_[n] = s * att_src[0];
    ad_[n] = s * att_dst[0];
}

// ---------------- scalar aggregation (last layer, C=1) ----------------------
__global__ void edge_agg1(const int* __restrict__ src, const int* __restrict__ dst,
                          int E0, int Etot, const float* __restrict__ ex,
                          const float* __restrict__ den, const float* __restrict__ h1,
                          float* __restrict__ out)
{
    int e = blockIdx.x * blockDim.x + threadIdx.x;
    if (e >= Etot) return;
    int s, d;
    if (e < E0) { s = src[e]; d = dst[e]; } else { s = d = e - E0; }
    atomicAdd(&out[d], (ex[e] / den[d]) * h1[s]);
}

__global__ void add_bias1(float* __restrict__ o, const float* __restrict__ bias, int M)
{
    int n = blockIdx.x * blockDim.x + threadIdx.x;
    if (n < M) o[n] += bias[0];
}

// ===========================================================================
extern "C" void kernel_launch(void* const* d_in, const int* in_sizes, int n_in,
                              void* d_out, int out_size, void* d_ws, size_t ws_size,
                              hipStream_t stream)
{
    (void)n_in; (void)out_size; (void)ws_size;

    const int M    = in_sizes[0] / CDIM;     // 80000 nodes
    const int E0   = in_sizes[2];            // 1280000 edges
    const int Etot = E0 + M;                 // + self loops

    const float* x  = (const float*)d_in[0];
    const int*   ei = (const int*)d_in[1];   // [2, E0] int32
    const float* ea = (const float*)d_in[2];
    const int* srcI = ei;
    const int* dstI = ei + E0;
    float* out = (float*)d_out;

    // ---- workspace layout (floats) --------------------------------------
    float*    ws    = (float*)d_ws;
    float*    H     = ws;                       // [M,128] transformed feats
    float*    O     = H + (size_t)M * CDIM;     // [M,128] aggregation / next input
    float*    AS    = O + (size_t)M * CDIM;     // [M]
    float*    AD    = AS + M;                   // [M]
    unsigned* MX    = (unsigned*)(AD + M);      // [M] encoded segment max
    float*    DEN   = (float*)(MX + M);         // [M]
    float*    SCORE = DEN + M;                  // [Etot] score -> exp
    float*    SC    = SCORE + Etot;             // scalars: [0]=sum(ea), [1]=k_e
    // total ~= 22.2M floats (~89 MB)

    // mean edge attribute (numerator) for self loops
    hipMemsetAsync(SC, 0, sizeof(float), stream);
    sum_reduce<<<256, 256, 0, stream>>>(ea, E0, SC);

    const int tpb = 256;
    const int nodeBlocks = (M + tpb - 1) / tpb;
    const int edgeBlocks = (Etot + tpb - 1) / tpb;
    const int aggBlocks  = (Etot + 7) / 8;
    const int gemmBlocks = (M / 16 + 7) / 8;    // 8 row-stripes (waves) per block

    // pytree leaves assumed in JAX sorted-key order per layer:
    //   W, We, att_dst, att_edge, att_src, bias
    for (int net = 0; net < 2; ++net) {
        float* finalOut = out + (size_t)net * M;
        for (int layer = 0; layer < 3; ++layer) {
            const int p = 3 + net * 18 + layer * 6;
            const float* W        = (const float*)d_in[p + 0];
            const float* We       = (const float*)d_in[p + 1];
            const float* att_dst  = (const float*)d_in[p + 2];
            const float* att_edge = (const float*)d_in[p + 3];
            const float* att_src  = (const float*)d_in[p + 4];
            const float* bias     = (const float*)d_in[p + 5];

            if (layer < 2) {
                const float* hin = (layer == 0) ? x : O;
                gemm_wmma_f32<<<gemmBlocks, 256, GEMM_LDS_BYTES, stream>>>(hin, W, H, M);
                node_dots<<<nodeBlocks, tpb, 0, stream>>>(H, att_src, att_dst, AS, AD, M);
                dot_small<<<1, 128, 0, stream>>>(We, att_edge, CDIM, SC + 1);
            } else {
                matvec_last<<<nodeBlocks, tpb, 0, stream>>>(O, W, att_src, att_dst,
                                                            H, AS, AD, M);
                dot_small<<<1, 128, 0, stream>>>(We, att_edge, 1, SC + 1);
            }

            hipMemsetAsync(MX, 0, (size_t)M * 4, stream);   // fenc init ~ -inf
            edge_score<<<edgeBlocks, tpb, 0, stream>>>(srcI, dstI, ea, E0, Etot,
                                                       AS, AD, SC + 1, SC, SCORE, MX);

            hipMemsetAsync(DEN, 0, (size_t)M * 4, stream);
            edge_exp<<<edgeBlocks, tpb, 0, stream>>>(dstI, E0, Etot, SCORE, MX, DEN);

            if (layer < 2) {
                hipMemsetAsync(O, 0, (size_t)M * CDIM * 4, stream);
                edge_agg128<<<aggBlocks, 256, 0, stream>>>(srcI, dstI, E0, Etot,
                                                           SCORE, DEN, H, O);
                post_act<<<(M * CDIM + tpb - 1) / tpb, tpb, 0, stream>>>(O, bias, M * CDIM);
            } else {
                hipMemsetAsync(finalOut, 0, (size_t)M * 4, stream);
                edge_agg1<<<edgeBlocks, tpb, 0, stream>>>(srcI, dstI, E0, Etot,
                                                          SCORE, DEN, H, finalOut);
                add_bias1<<<nodeBlocks, tpb, 0, stream>>>(finalOut, bias, M);
            }
        }
    }
}